// GNN_Att_Layer_31284541784231
// MI455X (gfx1250) — compile-verified
//
#include <hip/hip_runtime.h>

typedef float v2f __attribute__((ext_vector_type(2)));
typedef float v8f __attribute__((ext_vector_type(8)));

#define NB   2048
#define DM   256
#define NHEAD 8
#define DHEAD 32

__device__ __forceinline__ v8f wmma4(v2f a, v2f b, v8f c) {
  // V_WMMA_F32_16X16X4_F32 : D = A(16x4,f32) x B(4x16,f32) + C(16x16,f32)
  return __builtin_amdgcn_wmma_f32_16x16x4_f32(false, a, false, b, (short)0, c,
                                               false, false);
}

// Butterfly max over each 16-lane half (DPP16 ROW_XMASK: pure VALU, no DS).
__device__ __forceinline__ float red16_max(float v) {
  int x;
  x = __builtin_amdgcn_update_dpp(0, __float_as_int(v), 0x161, 0xf, 0xf, true);
  v = fmaxf(v, __int_as_float(x));
  x = __builtin_amdgcn_update_dpp(0, __float_as_int(v), 0x162, 0xf, 0xf, true);
  v = fmaxf(v, __int_as_float(x));
  x = __builtin_amdgcn_update_dpp(0, __float_as_int(v), 0x164, 0xf, 0xf, true);
  v = fmaxf(v, __int_as_float(x));
  x = __builtin_amdgcn_update_dpp(0, __float_as_int(v), 0x168, 0xf, 0xf, true);
  v = fmaxf(v, __int_as_float(x));
  return v;
}

// ---------------------------------------------------------------------------
// Fused GEMM: out = act( X[M x 256] @ W[256 x 256] + bias (+ resid + rowvec) )
// MODE 0: plain   MODE 1: tanh   MODE 2: + resid + rowvec[b,:] then ELU
// One wave -> 16 rows x 64 cols. Block = 256 thr = 8 waves = 2 row tiles.
// ---------------------------------------------------------------------------
template <int MODE>
__global__ void __launch_bounds__(256)
gemm256(const float* __restrict__ X, const float* __restrict__ W,
        const float* __restrict__ bias, const float* __restrict__ resid,
        const float* __restrict__ rowvec, float* __restrict__ out) {
  const int tid  = threadIdx.x;
  const int wave = tid >> 5;
  const int lane = tid & 31;
  const int n16  = lane & 15;
  const int hi   = lane >> 4;

  const int rowtile = blockIdx.x * 2 + (wave >> 2);
  const int colbase = (wave & 3) * 64;
  const int row_a   = rowtile * 16 + n16;     // A-layout: M = lane&15

  const float* xrow = X + (size_t)row_a * DM + 2 * hi;
  const float* wp   = W + (size_t)(2 * hi) * DM + colbase + n16;

  v8f acc[4] = {};
#pragma unroll 4
  for (int k = 0; k < DM; k += 4) {
    v2f a = *(const v2f*)(xrow + k);                   // A[m][k+2hi .. +1]
#pragma unroll
    for (int c = 0; c < 4; ++c) {
      v2f b;
      b.x = wp[(size_t)k * DM + c * 16];       // B[k+2hi][n]
      b.y = wp[(size_t)k * DM + c * 16 + DM];  // B[k+2hi+1][n]
      acc[c] = wmma4(a, b, acc[c]);
    }
  }

#pragma unroll
  for (int c = 0; c < 4; ++c) {
    const int col = colbase + c * 16 + n16;
    const float bv = bias[col];
#pragma unroll
    for (int r = 0; r < 8; ++r) {
      const int row = rowtile * 16 + r + 8 * hi;   // C-layout row
      float v = acc[c][r] + bv;
      if (MODE == 1) {
        v = tanhf(v);
      } else if (MODE == 2) {
        v += resid[(size_t)row * DM + col];
        v += rowvec[(size_t)(row >> 11) * DM + col];   // row/2048 = batch
        v = (v > 0.0f) ? v : expm1f(v);                // ELU
      }
      out[(size_t)row * DM + col] = v;
    }
  }
}

// ---------------------------------------------------------------------------
// Streaming (flash-style) graph-masked attention.
// One wave per (b, head, 16-query tile); loops over 128 key tiles of 16.
// Row-max via DPP; running softmax denominator via WMMA against all-ones B.
// All global streams use a single bumped base pointer + imm24 offsets.
// ---------------------------------------------------------------------------
__global__ void __launch_bounds__(256)
attn_kernel(const float* __restrict__ Q, const float* __restrict__ K,
            const float* __restrict__ V, const int* __restrict__ adj,
            float* __restrict__ ctx) {
  __shared__ float sP[8][16 * 18];   // per-wave P tile, stride 18 (8B aligned)

  const int tid  = threadIdx.x;
  const int wave = tid >> 5;
  const int lane = tid & 31;
  const int n16  = lane & 15;
  const int hi   = lane >> 4;

  const int gw = blockIdx.x * 8 + wave;
  const int qt = gw & 127;          // query tile
  const int h  = (gw >> 7) & 7;     // head
  const int b  = gw >> 10;          // batch

  // Preload the 8 A-fragments of Q (16 x 32 per head), pre-scaled by 1/sqrt(Dh).
  const float scale = 0.17677669529663687f;   // 1/sqrt(32)
  const float* qbase =
      Q + (size_t)(b * NB + qt * 16 + n16) * DM + h * DHEAD;
  v2f aQ[8];
#pragma unroll
  for (int j = 0; j < 8; ++j) {
    v2f q = *(const v2f*)(qbase + 4 * j + 2 * hi);
    q.x *= scale;
    q.y *= scale;
    aQ[j] = q;
  }

  v8f o0 = {}, o1 = {}, lacc = {};
  float mrow[8];
#pragma unroll
  for (int r = 0; r < 8; ++r) mrow[r] = -3.0e38f;

  // Bumped base pointers; all per-row/fragment deltas are imm24 offsets.
  const int*   adjp = adj + ((size_t)(b * NB + qt * 16 + 8 * hi)) * NB + n16;
  const float* kp   = K + ((size_t)b * NB + n16) * DM + h * DHEAD + 2 * hi;
  const float* vp   = V + ((size_t)b * NB + 2 * hi) * DM + h * DHEAD + n16;

  float* sp = &sP[wave][0];
  const v2f ones = {1.0f, 1.0f};

  for (int kt = 0; kt < 128; ++kt) {
    // Prefetch next adjacency tile (adj is the single real HBM stream).
    __builtin_prefetch(adjp + 16, 0, 0);

    // ---- S = (Q/sqrt(Dh)) @ K^T for this 16x16 tile (8 WMMAs) ----
    v8f s = {};
#pragma unroll
    for (int j = 0; j < 8; ++j) {
      v2f bK = *(const v2f*)(kp + 4 * j);            // B[k][n]=K[key n][k]
      s = wmma4(aQ[j], bK, s);
    }
    kp += 16 * DM;

    // ---- mask with adjacency (8 loads at r*8192B imm offsets) ----
#pragma unroll
    for (int r = 0; r < 8; ++r) {
      const int a = adjp[(size_t)r * NB];
      s[r] = a ? s[r] : -1.0e9f;
    }
    adjp += 16;

    // ---- online softmax: DPP row-max, exp, rescale running state ----
#pragma unroll
    for (int r = 0; r < 8; ++r) {
      const float mnew = fmaxf(mrow[r], red16_max(s[r]));
      const float corr = __expf(mrow[r] - mnew);
      mrow[r] = mnew;
      s[r] = __expf(s[r] - mnew);
      o0[r] *= corr;
      o1[r] *= corr;
      lacc[r] *= corr;
    }

    // ---- reshape P: C-layout -> A-layout via wave-private LDS ----
#pragma unroll
    for (int r = 0; r < 8; ++r) sp[(r + 8 * hi) * 18 + n16] = s[r];
    // per-wave DS ops are in order; compiler inserts s_wait_dscnt.

    // ---- O += P @ V_tile; denominator lacc += P @ Ones (row sums) ----
#pragma unroll
    for (int j = 0; j < 4; ++j) {
      v2f aP = *(const v2f*)(sp + n16 * 18 + 4 * j + 2 * hi);
      v2f b0, b1;
      b0.x = vp[(size_t)(4 * j) * DM];             // V[key][d]
      b0.y = vp[(size_t)(4 * j + 1) * DM];         // V[key+1][d]
      b1.x = vp[(size_t)(4 * j) * DM + 16];
      b1.y = vp[(size_t)(4 * j + 1) * DM + 16];
      o0   = wmma4(aP, b0, o0);
      o1   = wmma4(aP, b1, o1);
      lacc = wmma4(aP, ones, lacc);
    }
    vp += 16 * DM;
  }

  // ---- normalize and store ctx[b, q, h*32 + d] ----
  float* cb = ctx + (size_t)(b * NB + qt * 16) * DM + h * DHEAD;
#pragma unroll
  for (int r = 0; r < 8; ++r) {
    const float inv = 1.0f / lacc[r];
    const int qrow = r + 8 * hi;
    cb[(size_t)qrow * DM + n16]      = o0[r] * inv;
    cb[(size_t)qrow * DM + 16 + n16] = o1[r] * inv;
  }
}

// ---------------------------------------------------------------------------
// Masked mean pool of G over sequence dim: hg_pre[b,d] = sum_n G*mf / sum_n mf
// ---------------------------------------------------------------------------
__global__ void __launch_bounds__(256)
pool_kernel(const float* __restrict__ G, const unsigned char* __restrict__ mask,
            float* __restrict__ hg_pre) {
  const int b = blockIdx.x;
  const int d = threadIdx.x;
  __shared__ float red[256];

  float cnt = 0.0f;
  for (int n = d; n < NB; n += 256) cnt += mask[b * NB + n] ? 1.0f : 0.0f;
  red[d] = cnt;
  __syncthreads();
  for (int s = 128; s > 0; s >>= 1) {
    if (d < s) red[d] += red[d + s];
    __syncthreads();
  }
  const float len = red[0];

  float acc = 0.0f;
  const float* gb = G + (size_t)b * NB * DM + d;
  const unsigned char* mb = mask + b * NB;
  for (int n = 0; n < NB; ++n)
    if (mb[n]) acc += gb[(size_t)n * DM];
  hg_pre[b * DM + d] = acc / len;
}

// ---------------------------------------------------------------------------
// Tiny GEMM: hg[b,:] = hg_pre[b,:] @ Wgo + bgo   ([4x256]x[256x256])
// ---------------------------------------------------------------------------
__global__ void __launch_bounds__(256)
global_node_kernel(const float* __restrict__ hg_pre,
                   const float* __restrict__ Wgo, const float* __restrict__ bgo,
                   float* __restrict__ hg) {
  const int b = blockIdx.x;
  const int d = threadIdx.x;
  float acc = bgo[d];
  const float* hb = hg_pre + b * DM;
  for (int k = 0; k < DM; ++k) acc += hb[k] * Wgo[k * DM + d];
  hg[b * DM + d] = acc;
}

// ---------------------------------------------------------------------------
extern "C" void kernel_launch(void* const* d_in, const int* in_sizes, int n_in,
                              void* d_out, int out_size, void* d_ws,
                              size_t ws_size, hipStream_t stream) {
  const float* x   = (const float*)d_in[0];
  const int*   adj = (const int*)d_in[1];
  const unsigned char* mask = (const unsigned char*)d_in[2];
  const float* Wq = (const float*)d_in[3];  const float* bq = (const float*)d_in[4];
  const float* Wk = (const float*)d_in[5];  const float* bk = (const float*)d_in[6];
  const float* Wv = (const float*)d_in[7];  const float* bv = (const float*)d_in[8];
  const float* Wo = (const float*)d_in[9];  const float* bo = (const float*)d_in[10];
  const float* Wl = (const float*)d_in[11]; const float* bl = (const float*)d_in[12];
  const float* Wgi = (const float*)d_in[13]; const float* bgi = (const float*)d_in[14];
  const float* Wgo = (const float*)d_in[15]; const float* bgo = (const float*)d_in[16];

  float* out = (float*)d_out;
  float* ws  = (float*)d_ws;

  const size_t S = (size_t)4 * NB * DM;   // 2,097,152 floats per tensor
  float* Qb = ws;          float* Kb = ws + S;      float* Vb = ws + 2 * S;
  float* Lb = ws + 3 * S;  float* Gb = ws + 4 * S;  float* CTX = ws + 5 * S;
  float* HGP = ws + 6 * S; float* HG = ws + 6 * S + 1024;

  dim3 blk(256);
  // Stage 1: five fused projections (WMMA fp32 GEMMs)
  gemm256<0><<<256, blk, 0, stream>>>(x, Wq, bq, nullptr, nullptr, Qb);
  gemm256<0><<<256, blk, 0, stream>>>(x, Wk, bk, nullptr, nullptr, Kb);
  gemm256<0><<<256, blk, 0, stream>>>(x, Wv, bv, nullptr, nullptr, Vb);
  gemm256<0><<<256, blk, 0, stream>>>(x, Wl, bl, nullptr, nullptr, Lb);
  gemm256<1><<<256, blk, 0, stream>>>(x, Wgi, bgi, nullptr, nullptr, Gb);

  // Stage 2: streaming graph-masked attention (4096 waves)
  attn_kernel<<<512, blk, 0, stream>>>(Qb, Kb, Vb, adj, CTX);

  // Stage 3: global node pooling + tiny GEMM
  pool_kernel<<<4, blk, 0, stream>>>(Gb, mask, HGP);
  global_node_kernel<<<4, blk, 0, stream>>>(HGP, Wgo, bgo, HG);

  // Stage 4: out = ELU(ctx@Wo + bo + L + hg[b,:])
  gemm256<2><<<256, blk, 0, stream>>>(CTX, Wo, bo, Lb, HG, out);
}